// Softmax_62397284876677
// MI455X (gfx1250) — compile-verified
//
#include <hip/hip_runtime.h>

// ---------------------------------------------------------------------------
// softmax + double softmax-Jacobian application, B=8192 rows, C=8192 cols fp32
// d_out = [ s (B*C floats) | sigma_out (B*C floats) ]
// Bandwidth-bound: ~1.07 GB min traffic -> ~46us floor at 23.3 TB/s.
// CDNA5 path: async global->LDS staging of sigma (ASYNCcnt), wave32 shuffles,
// B128 everything, NT hints to keep write-once streams out of L2.
// ---------------------------------------------------------------------------

#define CCOLS 8192
#define TPB   256
#define CHUNK 8   // CCOLS / (TPB*4)

typedef float v4f __attribute__((ext_vector_type(4)));
typedef int   v4i __attribute__((vector_size(16)));
typedef __attribute__((address_space(1))) v4i* gptr_v4i;
typedef __attribute__((address_space(3))) v4i* lptr_v4i;

#if defined(__has_builtin)
#if __has_builtin(__builtin_amdgcn_global_load_async_to_lds_b128)
#define HAVE_ASYNC_LDS 1
#endif
#if __has_builtin(__builtin_amdgcn_s_wait_asynccnt)
#define HAVE_WAIT_ASYNC 1
#endif
#endif

__device__ __forceinline__ void stage16_to_lds(const float* g, float* l) {
#if defined(HAVE_ASYNC_LDS)
  // async copy 16B/lane global -> LDS, tracked with ASYNCcnt (no VGPR data)
  __builtin_amdgcn_global_load_async_to_lds_b128(
      (gptr_v4i)(g), (lptr_v4i)(l), 0, 0);
#else
  *(v4f*)l = __builtin_nontemporal_load((const v4f*)g);
#endif
}

__device__ __forceinline__ void wait_async_all() {
#if defined(HAVE_ASYNC_LDS)
#if defined(HAVE_WAIT_ASYNC)
  __builtin_amdgcn_s_wait_asynccnt(0);
#else
  asm volatile("s_wait_asynccnt 0" ::: "memory");
#endif
#endif
}

__device__ __forceinline__ float waveRedMax(float v) {
#pragma unroll
  for (int m = 16; m; m >>= 1) v = fmaxf(v, __shfl_xor(v, m, 32));
  return v;
}

__device__ __forceinline__ float waveRedSum(float v) {
#pragma unroll
  for (int m = 16; m; m >>= 1) v += __shfl_xor(v, m, 32);
  return v;
}

__global__ __launch_bounds__(TPB) void Softmax_62397284876677_kernel(
    const float* __restrict__ mu, const float* __restrict__ sg,
    float* __restrict__ outS, float* __restrict__ outD) {
  __shared__ float sgL[CCOLS];   // 32 KB sigma staging
  __shared__ float red[24];      // cross-wave reduction scratch

  const int t    = threadIdx.x;
  const int lane = t & 31;
  const int wid  = t >> 5;
  const size_t rb = (size_t)blockIdx.x * CCOLS;
  const float* gmu = mu + rb;
  const float* gsg = sg + rb;

  // ---- kick off async staging of the sigma row; overlaps phases A+B -------
#pragma unroll
  for (int c = 0; c < CHUNK; ++c) {
    const int off = c * (TPB * 4) + t * 4;
    stage16_to_lds(gsg + off, &sgL[off]);
  }

  // ---- phase A: stream mu straight to VGPRs (read-once -> NT), row max ----
  v4f e[CHUNK];
  float mx = -3.402823466e+38f;
#pragma unroll
  for (int c = 0; c < CHUNK; ++c) {
    e[c] = __builtin_nontemporal_load((const v4f*)(gmu + c * (TPB * 4) + t * 4));
    mx = fmaxf(mx, fmaxf(fmaxf(e[c].x, e[c].y), fmaxf(e[c].z, e[c].w)));
  }
  mx = waveRedMax(mx);
  if (lane == 0) red[wid] = mx;
  __syncthreads();
  float rmax = red[0];
#pragma unroll
  for (int i = 1; i < 8; ++i) rmax = fmaxf(rmax, red[i]);
  __syncthreads();  // red about to be reused

  // ---- phase B: exp in registers, row sum ---------------------------------
  float z = 0.f;
#pragma unroll
  for (int c = 0; c < CHUNK; ++c) {
    v4f v = e[c];
    v.x = __expf(v.x - rmax);
    v.y = __expf(v.y - rmax);
    v.z = __expf(v.z - rmax);
    v.w = __expf(v.w - rmax);
    e[c] = v;
    z += (v.x + v.y) + (v.z + v.w);
  }
  z = waveRedSum(z);
  if (lane == 0) red[wid] = z;
  __syncthreads();
  float zt = 0.f;
#pragma unroll
  for (int i = 0; i < 8; ++i) zt += red[i];
  __syncthreads();  // red reused for 3-way reduction below
  const float invZ = 1.0f / zt;

  // ---- sigma must be resident now -----------------------------------------
  wait_async_all();   // this wave's async LDS writes done
  __syncthreads();    // all waves' writes visible

  // ---- phase C: s = e/Z, write s, fused A=Σs·σ, B=Σs²·σ, C2=Σs² ----------
  float a = 0.f, b = 0.f, cc = 0.f;
#pragma unroll
  for (int c = 0; c < CHUNK; ++c) {
    const int off = c * (TPB * 4) + t * 4;
    const v4f sig = *(const v4f*)&sgL[off];
    const v4f s   = e[c] * invZ;
    e[c] = s;  // e now holds s
    __builtin_nontemporal_store(s, (v4f*)(outS + rb + off));
    const v4f ss = s * sig;
    const v4f s2 = s * s;
    a  += (ss.x + ss.y) + (ss.z + ss.w);
    cc += (s2.x + s2.y) + (s2.z + s2.w);
    const v4f s2s = s2 * sig;
    b  += (s2s.x + s2s.y) + (s2s.z + s2s.w);
  }
  a  = waveRedSum(a);
  b  = waveRedSum(b);
  cc = waveRedSum(cc);
  if (lane == 0) { red[wid] = a; red[8 + wid] = b; red[16 + wid] = cc; }
  __syncthreads();
  float A = 0.f, Bv = 0.f, Cv = 0.f;
#pragma unroll
  for (int i = 0; i < 8; ++i) { A += red[i]; Bv += red[8 + i]; Cv += red[16 + i]; }
  const float t2 = Bv - A * Cv;  // = sum(s * jvp)

  // ---- phase D: sigma_out = s*(s*(sigma - A) - t2) ------------------------
#pragma unroll
  for (int c = 0; c < CHUNK; ++c) {
    const int off = c * (TPB * 4) + t * 4;
    const v4f sig = *(const v4f*)&sgL[off];
    const v4f s   = e[c];
    const v4f so  = s * (s * (sig - A) - t2);
    __builtin_nontemporal_store(so, (v4f*)(outD + rb + off));
  }
}

extern "C" void kernel_launch(void* const* d_in, const int* in_sizes, int n_in,
                              void* d_out, int out_size, void* d_ws, size_t ws_size,
                              hipStream_t stream) {
  const float* mu = (const float*)d_in[0];
  const float* sg = (const float*)d_in[1];
  float* out = (float*)d_out;
  const int rows = in_sizes[0] / CCOLS;
  float* outS = out;
  float* outD = out + (size_t)rows * CCOLS;
  Softmax_62397284876677_kernel<<<dim3(rows), dim3(TPB), 0, stream>>>(mu, sg, outS, outD);
}